// FullyConnected_76605036691670
// MI455X (gfx1250) — compile-verified
//
#include <hip/hip_runtime.h>

// MAM linear layer: C[n,m] = max_k(x[n,k]*w[m,k]) + min_k(...) + bias[m]
// plus argmax/argmin over k. N=1024 (8*128), K=512, M=512, all f32.
//
// CDNA5 strategy: TDM (tensor_load_to_lds) double-buffered LDS pipeline for
// x/w tiles (async, TENSORcnt-tracked), VALU compare/select reduction.

#define N_TOT 1024
#define K_TOT 512
#define M_TOT 512
#define BN 32          // rows of x per block
#define BM 64          // rows of w (output cols) per block
#define KC 64          // K-chunk per TDM load
#define KSTR (KC + 1)  // LDS row stride in dwords (TDM pad: +1 dword / 64 dwords)
#define NCHUNK (K_TOT / KC)

typedef unsigned int u32x4 __attribute__((ext_vector_type(4)));
typedef int          i32x4 __attribute__((ext_vector_type(4)));
typedef int          i32x8 __attribute__((ext_vector_type(8)));

// Build a 2-D tile TDM descriptor and issue TENSOR_LOAD_TO_LDS.
// Tensor: [tensor_d1 rows x tensor_d0 cols] f32, row stride d0_stride elems.
// Tile:   [tile_d1 rows x tile_d0 cols] starting at gaddr, landing at lds_addr
// with 1 dword of padding inserted after every 64 dwords (pad_interval=5),
// giving an LDS row stride of KSTR dwords (bank-conflict-free reads).
__device__ __forceinline__ void tdm_load_2d(unsigned lds_addr,
                                            unsigned long long gaddr,
                                            unsigned tensor_d0,
                                            unsigned tensor_d1,
                                            unsigned long long d0_stride,
                                            unsigned tile_d0,
                                            unsigned tile_d1) {
  u32x4 g0;
  g0[0] = 1u;                                          // count=1 (valid), user mode
  g0[1] = lds_addr;                                    // LDS byte address
  g0[2] = (unsigned)(gaddr & 0xFFFFFFFFull);           // global_addr[31:0]
  g0[3] = (unsigned)((gaddr >> 32) & 0x01FFFFFFull)    // global_addr[56:32]
          | (2u << 30);                                // type=2 ("image")

  i32x8 g1;
  // data_size=2 (4B) | pad_enable | pad_interval=5 (64 dwords) | pad_amount=0 (1 dword)
  g1[0] = (int)((2u << 16) | (1u << 20) | (5u << 22));
  g1[1] = (int)((tensor_d0 & 0xFFFFu) << 16);          // [47:32]=atomic_barrier=0, [63:48]=dim0 lo16
  g1[2] = (int)(((tensor_d0 >> 16) & 0xFFFFu) |        // dim0 hi16
                ((tensor_d1 & 0xFFFFu) << 16));        // dim1 lo16
  g1[3] = (int)(((tensor_d1 >> 16) & 0xFFFFu) |        // dim1 hi16
                ((tile_d0 & 0xFFFFu) << 16));          // tile_dim0
  g1[4] = (int)(tile_d1 & 0xFFFFu);                    // tile_dim1 (tile_dim2=0)
  g1[5] = (int)(unsigned)(d0_stride & 0xFFFFFFFFull);  // dim0_stride lo32
  g1[6] = (int)(unsigned)((d0_stride >> 32) & 0xFFFFull); // dim0_stride hi16, dim1_stride=0
  g1[7] = 0;

  i32x4 z4 = {0, 0, 0, 0};                             // groups 2/3 unused (2-D)
  i32x8 z8 = {0, 0, 0, 0, 0, 0, 0, 0};                 // extra group (6-arg toolchain)
  __builtin_amdgcn_tensor_load_to_lds(g0, g1, z4, z4, z8, 0);
}

__global__ __launch_bounds__(256) void mam_fc_kernel(
    const float* __restrict__ x, const float* __restrict__ w,
    const float* __restrict__ bias, float* __restrict__ out) {
  __shared__ float lx[2][BN][KSTR];   // 2 x 32 x 65 f32 = 16.25 KB
  __shared__ float lw[2][BM][KSTR];   // 2 x 64 x 65 f32 = 32.5 KB

  const int tid = threadIdx.x;
  const int nb  = blockIdx.x;   // 0..31  (n tiles)
  const int mb  = blockIdx.y;   // 0..7   (m tiles)
  const int tmg = tid & 15;     // m group: owns m = tmg + 16*j, j=0..3
  const int tng = tid >> 4;     // n group: owns n = tng, tng+16

  float vmax[2][4], vmin[2][4];
  int   imax[2][4], imin[2][4];
#pragma unroll
  for (int i = 0; i < 2; ++i)
#pragma unroll
    for (int j = 0; j < 4; ++j) {
      vmax[i][j] = -__builtin_inff();
      vmin[i][j] =  __builtin_inff();
      imax[i][j] = 0;
      imin[i][j] = 0;
    }

  auto issue = [&](int chunk, int buf) {
    unsigned long long gx = (unsigned long long)(const void*)x +
        ((unsigned long long)(nb * BN) * K_TOT + (unsigned long long)chunk * KC) * 4ull;
    unsigned long long gw = (unsigned long long)(const void*)w +
        ((unsigned long long)(mb * BM) * K_TOT + (unsigned long long)chunk * KC) * 4ull;
    tdm_load_2d((unsigned)(size_t)&lx[buf][0][0], gx, K_TOT, N_TOT, K_TOT, KC, BN);
    tdm_load_2d((unsigned)(size_t)&lw[buf][0][0], gw, K_TOT, M_TOT, K_TOT, KC, BM);
  };

  // Prologue: wave 0 kicks off chunk 0 into buffer 0 (TENSORcnt = 2).
  if (tid < 32) issue(0, 0);

  int buf = 0;
  for (int c = 0; c < NCHUNK; ++c) {
    if (tid < 32) {
      if (c + 1 < NCHUNK) {
        issue(c + 1, buf ^ 1);                 // prefetch next chunk (cnt -> 4)
        __builtin_amdgcn_s_wait_tensorcnt(2);  // chunk c's two loads done
      } else {
        __builtin_amdgcn_s_wait_tensorcnt(0);  // last chunk: drain
      }
    }
    __syncthreads();  // chunk c visible in LDS to all waves

    const int kbase = c * KC;
#pragma unroll 4
    for (int kk = 0; kk < KC; ++kk) {
      const int kg = kbase + kk;
      const float xa = lx[buf][tng][kk];
      const float xb = lx[buf][tng + 16][kk];
      float wv[4];
#pragma unroll
      for (int j = 0; j < 4; ++j) wv[j] = lw[buf][tmg + 16 * j][kk];
#pragma unroll
      for (int j = 0; j < 4; ++j) {
        const float pa = xa * wv[j];
        const float pb = xb * wv[j];
        if (pa > vmax[0][j]) { vmax[0][j] = pa; imax[0][j] = kg; }
        if (pa < vmin[0][j]) { vmin[0][j] = pa; imin[0][j] = kg; }
        if (pb > vmax[1][j]) { vmax[1][j] = pb; imax[1][j] = kg; }
        if (pb < vmin[1][j]) { vmin[1][j] = pb; imin[1][j] = kg; }
      }
    }
    __syncthreads();  // all waves done reading buf before it is overwritten
    buf ^= 1;
  }

  // Epilogue: C (f32), argmax (i32), argmin (i32), concatenated flat.
  float* outc = out;
  int* outamax = (int*)out + (size_t)N_TOT * M_TOT;
  int* outamin = (int*)out + 2ull * (size_t)N_TOT * M_TOT;
#pragma unroll
  for (int i = 0; i < 2; ++i) {
    const int n = nb * BN + tng + 16 * i;
#pragma unroll
    for (int j = 0; j < 4; ++j) {
      const int m = mb * BM + tmg + 16 * j;
      const size_t off = (size_t)n * M_TOT + m;
      outc[off]    = vmax[i][j] + vmin[i][j] + bias[m];
      outamax[off] = imax[i][j];
      outamin[off] = imin[i][j];
    }
  }
}

extern "C" void kernel_launch(void* const* d_in, const int* in_sizes, int n_in,
                              void* d_out, int out_size, void* d_ws, size_t ws_size,
                              hipStream_t stream) {
  (void)in_sizes; (void)n_in; (void)out_size; (void)d_ws; (void)ws_size;
  const float* x    = (const float*)d_in[0];
  const float* wgt  = (const float*)d_in[1];
  const float* bias = (const float*)d_in[2];
  dim3 grid(N_TOT / BN, M_TOT / BM);  // 32 x 8 = 256 blocks
  mam_fc_kernel<<<grid, 256, 0, stream>>>(x, wgt, bias, (float*)d_out);
}